// TtS_24764781428908
// MI455X (gfx1250) — compile-verified
//
#include <hip/hip_runtime.h>
#include <cstdint>
#include <cstddef>

// Problem constants (from reference setup_inputs)
#define CCH   64
#define TT    32
#define HH    56
#define WW    56
#define HWSZ  (HH*WW)

// Tiling
#define TILE_H   8
#define NROWS    10            // TILE_H + 2 halo rows
#define RWORDS   64            // LDS row stride in words: 56 data @ word 4, halos @ 3 and 60
#define SLOTW    (NROWS*RWORDS) // 640 words per frame slot
#define NSLOT    9             // temporal ring depth
#define NTHREADS 448           // 14 wave32, one pixel per thread
#define NTILES_H 7             // HH / TILE_H
#define NLOADERS 160           // waves 0..4 issue async loads (wave-aligned)
#define NREAL    140           // 10 rows * 14 B128-groups per frame tile
#define DUMMYW   256           // 64 lanes * 4-word (16B) dummy sinks

__device__ __forceinline__ float fast_sigmoid(float x) {
  return 1.0f / (1.0f + __expf(-x));   // v_exp_f32 + v_rcp path
}

__global__ __launch_bounds__(NTHREADS)
void tgate_kernel(const float* __restrict__ in,
                  const float* __restrict__ w1, const float* __restrict__ b1,
                  const float* __restrict__ w2, const float* __restrict__ b2,
                  const float* __restrict__ w3, const float* __restrict__ b3,
                  const float* __restrict__ wts,
                  float* __restrict__ out)
{
  alignas(16) __shared__ float smem[NSLOT*SLOTW + DUMMYW];

  const int tid = threadIdx.x;
  const int bc  = blockIdx.x / NTILES_H;          // (b*C + c)
  const int ht  = blockIdx.x - bc*NTILES_H;
  const int h0  = ht * TILE_H;
  const int c   = bc & (CCH - 1);

  // ---- zero all LDS: provides T<0 frames, H-edge halo rows, W halo cols.
  // Validity of each LDS position is t-invariant, so these zeros persist. ----
  for (int i = tid; i < NSLOT*SLOTW + DUMMYW; i += NTHREADS) smem[i] = 0.0f;

  // ---- per-channel weights (block-uniform -> scalar loads) ----
  float W1[27], W2[27], W3[27];
  #pragma unroll
  for (int i = 0; i < 27; ++i) {
    W1[i] = w1[c*27 + i];
    W2[i] = w2[c*27 + i];
    W3[i] = w3[c*27 + i];
  }
  const float B1 = b1[c], B2 = b2[c], B3 = b3[c];
  const float G0 = wts[0], G1 = wts[1], G2 = wts[2];

  // ---- per-thread output pixel ----
  const int hy = tid / WW;              // 0..7
  const int wx = tid - hy*WW;           // 0..55
  const int bi = hy*RWORDS + wx + 3;    // top-left of 3x3 window in a slot

  // ---- async B128 loader assignment (threads 0..159 only; wave-aligned so
  // ASYNCcnt is uniform per wave: waves 5..13 never issue, wait(0) is a nop).
  // Invalid lanes are address-redirected (same EXEC) to a 16B dummy sink and
  // all read the same frame-base 16B -> single coalesced L2 request. ----
  const uint32_t smem_base  = (uint32_t)(uintptr_t)(&smem[0]); // low 32 bits = LDS offset
  const uint32_t dummy_byte = smem_base + (uint32_t)((NSLOT*SLOTW + (tid & 63)*4) * 4);

  const int r   = tid / 14;             // row within tile (0..9 when real)
  const int g   = tid - r*14;           // 4-float group within row (0..13)
  const int grow = h0 - 1 + r;          // global H row
  const bool lval = (tid < NREAL) && (grow >= 0) && (grow < HH);
  const int  widx = r*RWORDS + 4 + 4*g;            // word index within slot
  const int  goff = lval ? (grow*WW + 4*g) : 0;    // element offset within frame

  const float* inbc = in + (size_t)bc * TT * HWSZ;

  auto issue_frame = [&](int f) {
    if (tid < NLOADERS) {
      const float* fb = inbc + (size_t)f * HWSZ;
      const uint32_t d = lval
          ? (smem_base + (uint32_t)(((f % NSLOT)*SLOTW + widx) * 4))
          : dummy_byte;
      const uint64_t a = (uint64_t)(uintptr_t)(fb + goff);
      asm volatile("global_load_async_to_lds_b128 %0, %1, off"
                   :: "v"(d), "v"(a)
                   : "memory");
    }
  };

  // zeros committed before any async load can overwrite a loaded position
  asm volatile("s_wait_dscnt 0x0" ::: "memory");
  __syncthreads();

  // preload frames 0..3 (slots 0..3); frames -3..-1 are the pre-zeroed slots 6,7,8
  issue_frame(0); issue_frame(1); issue_frame(2); issue_frame(3);

  // lag-diff history FIFOs
  float p1 = 0.f, p21 = 0.f, p22 = 0.f, p31 = 0.f, p32 = 0.f, p33 = 0.f;

  float* obc = out + (size_t)bc * TT * HWSZ + (size_t)(h0 + hy) * WW + wx;

  for (int t = 0; t < TT; ++t) {
    // frame t+3 (issued last iteration or in prologue) is complete after this
    asm volatile("s_wait_asynccnt 0x0" ::: "memory");
    __syncthreads();

    // prefetch frame t+4 into slot (t+4)%9 — not read by this step's compute,
    // and (ring depth 9) not read by any wave still finishing step t-1
    const int f = t + 4;
    if (f < TT) {
      issue_frame(f);
    } else if (f <= TT + 2) {          // frames 32..34 must read back as zero
      if (lval) {
        float4* p = reinterpret_cast<float4*>(&smem[(f % NSLOT)*SLOTW + widx]);
        *p = make_float4(0.f, 0.f, 0.f, 0.f);
      }
    }

    // ring slots for frames t-3 .. t+3
    int s[7];
    #pragma unroll
    for (int i = 0; i < 7; ++i) {
      int fr = t - 3 + i;
      int m  = fr % NSLOT; if (m < 0) m += NSLOT;
      s[i] = m;
    }

    // center frame window, read once, shared by all three convs (kt=1 taps)
    float xc[9];
    {
      const int sb = s[3]*SLOTW + bi;
      #pragma unroll
      for (int dh = 0; dh < 3; ++dh)
        #pragma unroll
        for (int dw = 0; dw < 3; ++dw)
          xc[dh*3 + dw] = smem[sb + dh*RWORDS + dw];
    }

    float y1 = B1, y2 = B2, y3 = B3;
    #pragma unroll
    for (int k = 0; k < 9; ++k) {
      y1 += xc[k]*W1[9 + k];
      y2 += xc[k]*W2[9 + k];
      y3 += xc[k]*W3[9 + k];
    }

    #define CONV9(slot, wp, acc) do {                                            \
        const int sb_ = (slot)*SLOTW + bi;                                       \
        acc += smem[sb_           ]*(wp)[0] + smem[sb_         +1]*(wp)[1]       \
             + smem[sb_         +2]*(wp)[2] + smem[sb_+RWORDS    ]*(wp)[3]       \
             + smem[sb_+RWORDS +1]*(wp)[4] + smem[sb_+RWORDS  +2]*(wp)[5]        \
             + smem[sb_+2*RWORDS ]*(wp)[6] + smem[sb_+2*RWORDS+1]*(wp)[7]        \
             + smem[sb_+2*RWORDS+2]*(wp)[8];                                     \
      } while (0)

    CONV9(s[2], W1 +  0, y1);   // frame t-1
    CONV9(s[4], W1 + 18, y1);   // frame t+1
    CONV9(s[1], W2 +  0, y2);   // frame t-2
    CONV9(s[5], W2 + 18, y2);   // frame t+2
    CONV9(s[0], W3 +  0, y3);   // frame t-3
    CONV9(s[6], W3 + 18, y3);   // frame t+3

    #undef CONV9

    // lag differences (first k frames pass through unchanged)
    const float d1 = (t >= 1) ? (y1 - p1)  : y1;
    const float d2 = (t >= 2) ? (y2 - p22) : y2;
    const float d3 = (t >= 3) ? (y3 - p33) : y3;
    p1 = y1;
    p22 = p21; p21 = y2;
    p33 = p32; p32 = p31; p31 = y3;

    const float gate = (fast_sigmoid(d1) - 0.5f) * G0
                     + (fast_sigmoid(d2) - 0.5f) * G1
                     + (fast_sigmoid(d3) - 0.5f) * G2;

    obc[(size_t)t * HWSZ] = xc[4] * gate;   // xc[4] == input center value
  }
}

extern "C" void kernel_launch(void* const* d_in, const int* in_sizes, int n_in,
                              void* d_out, int out_size, void* d_ws, size_t ws_size,
                              hipStream_t stream) {
  const float* in  = (const float*)d_in[0];
  const float* w1  = (const float*)d_in[1];
  const float* b1  = (const float*)d_in[2];
  const float* w2  = (const float*)d_in[3];
  const float* b2  = (const float*)d_in[4];
  const float* w3  = (const float*)d_in[5];
  const float* b3  = (const float*)d_in[6];
  const float* wts = (const float*)d_in[7];
  float* out = (float*)d_out;

  const int B = in_sizes[0] / (CCH * TT * HWSZ);   // = 4
  dim3 grid(B * CCH * NTILES_H);
  dim3 block(NTHREADS);
  tgate_kernel<<<grid, block, 0, stream>>>(in, w1, b1, w2, b2, w3, b3, wts, out);
}